// DRAELossAutograd_47880295415864
// MI455X (gfx1250) — compile-verified
//
#include <hip/hip_runtime.h>
#include <hip/hip_bf16.h>

#define LAMB 0.1f
constexpr int N_ROWS = 8192;
constexpr int D_COLS = 2048;

typedef __attribute__((ext_vector_type(2))) float v2f;
typedef __attribute__((ext_vector_type(8))) float v8f;

// ---------------------------------------------------------------------------
// Phase 1: err[row] = sum_k (in[row][k] - tg[row][k])^2
// One block = 16 rows. 8 waves, wave w owns columns [w*256, w*256+256).
// Per 8-column chunk: two b128 loads (double-buffered so 4 loads stay in
// flight), pk sub+square, two v_wmma_f32_16x16x4_f32 with B = ones.
// Because B is all-ones, D[m][n] = sum_k A[m][k] is invariant to which
// k-slot each value lands in, so no cross-lane shuffle is needed.
// ---------------------------------------------------------------------------
__global__ __launch_bounds__(256) void drae_err_kernel(const float* __restrict__ in,
                                                       const float* __restrict__ tg,
                                                       float* __restrict__ err) {
    __shared__ float lds_err[16];
    const int tid  = threadIdx.x;
    const int wave = tid >> 5;
    const int lane = tid & 31;
    const int m    = lane & 15;   // row within 16-row tile
    const int half = lane >> 4;   // A-matrix k-half (0: k=0,1  1: k=2,3)
    const int rowbase = blockIdx.x * 16;

    if (tid < 16) lds_err[tid] = 0.0f;
    __syncthreads();

    v8f acc = {0.f, 0.f, 0.f, 0.f, 0.f, 0.f, 0.f, 0.f};
    v2f bones; bones.x = 1.0f; bones.y = 1.0f;

    // per-lane base: row (rowbase+m), column (wave*256 + 4*half)
    const size_t lbase = (size_t)(rowbase + m) * D_COLS + wave * 256 + 4 * half;
    const float* pi = in + lbase;
    const float* pt = tg + lbase;

    // 2-deep software pipeline: keep 4 b128 loads outstanding
    float4 ivA = *(const float4*)(pi);
    float4 tvA = *(const float4*)(pt);
    float4 ivB = *(const float4*)(pi + 8);
    float4 tvB = *(const float4*)(pt + 8);

    #pragma unroll
    for (int it = 0; it < 32; ++it) {
        const float4 iv = ivA;
        const float4 tv = tvA;
        ivA = ivB; tvA = tvB;
        if (it + 2 < 32) {
            ivB = *(const float4*)(pi + (it + 2) * 8);
            tvB = *(const float4*)(pt + (it + 2) * 8);
        }
        float4 sq;
        { float dx = iv.x - tv.x; sq.x = dx * dx; }
        { float dy = iv.y - tv.y; sq.y = dy * dy; }
        { float dz = iv.z - tv.z; sq.z = dz * dz; }
        { float dw = iv.w - tv.w; sq.w = dw * dw; }
        v2f a0, a1;
        a0.x = sq.x; a0.y = sq.y;   // lane m: cols c,c+1   lane m+16: cols c+4,c+5
        a1.x = sq.z; a1.y = sq.w;   // lane m: cols c+2,c+3 lane m+16: cols c+6,c+7
        // D[m][n] += sum_k A[m][k] * 1 : every column n holds the row-sum
        acc = __builtin_amdgcn_wmma_f32_16x16x4_f32(false, a0, false, bones,
                                                    (short)0, acc, false, false);
        acc = __builtin_amdgcn_wmma_f32_16x16x4_f32(false, a1, false, bones,
                                                    (short)0, acc, false, false);
    }

    // C/D layout: lane 0 VGPR j -> row j (N=0); lane 16 VGPR j -> row 8+j (N=0)
    if (m == 0) {
        #pragma unroll
        for (int j = 0; j < 8; ++j)
            atomicAdd(&lds_err[half * 8 + j], acc[j]);   // ds_add_f32
    }
    __syncthreads();
    if (tid < 16) err[rowbase + tid] = lds_err[tid];
}

// ---------------------------------------------------------------------------
// Phase 2+3: single 1024-thread block.
// Bitonic sort of 8192 floats in LDS, dual prefix-scan (sum, sum^2),
// per-split within-class scatter, block argmin, final scalar.
// ---------------------------------------------------------------------------
__global__ __launch_bounds__(1024) void drae_sort_obj_kernel(const float* __restrict__ err,
                                                             float* __restrict__ out) {
    __shared__ float es[8192];
    __shared__ float s1[1024];
    __shared__ float s2[1024];
    __shared__ float hbuf[1024];
    __shared__ float c1buf[1024];
    __shared__ int   pbuf[1024];
    const int tid = threadIdx.x;

    for (int i = tid; i < N_ROWS; i += 1024) es[i] = err[i];
    __syncthreads();

    // bitonic sort ascending
    for (int k = 2; k <= N_ROWS; k <<= 1) {
        for (int j = k >> 1; j > 0; j >>= 1) {
            for (int i = tid; i < N_ROWS; i += 1024) {
                const int ixj = i ^ j;
                if (ixj > i) {
                    const float a = es[i], b = es[ixj];
                    const bool up = ((i & k) == 0);
                    if (up ? (a > b) : (a < b)) { es[i] = b; es[ixj] = a; }
                }
            }
            __syncthreads();
        }
    }

    // local inclusive scans over 8 owned consecutive elements
    float l1[8], l2[8];
    float r1 = 0.0f, r2 = 0.0f;
    const int p0 = tid * 8;
    #pragma unroll
    for (int q = 0; q < 8; ++q) {
        const float e = es[p0 + q];
        r1 += e; r2 += e * e;
        l1[q] = r1; l2[q] = r2;
    }
    s1[tid] = r1; s2[tid] = r2;
    __syncthreads();

    // Hillis-Steele inclusive scan over 1024 thread totals
    for (int off = 1; off < 1024; off <<= 1) {
        float a1 = s1[tid], a2 = s2[tid];
        if (tid >= off) { a1 += s1[tid - off]; a2 += s2[tid - off]; }
        __syncthreads();
        s1[tid] = a1; s2[tid] = a2;
        __syncthreads();
    }

    const float tot1 = s1[1023], tot2 = s2[1023];
    const float ex1 = (tid == 0) ? 0.0f : s1[tid - 1];
    const float ex2 = (tid == 0) ? 0.0f : s2[tid - 1];
    const float Nf = (float)N_ROWS;
    const float total_scatter = tot2 - tot1 * tot1 / Nf;

    float besth = INFINITY; int bestp = 0; float bestc1 = 0.0f;
    #pragma unroll
    for (int q = 0; q < 8; ++q) {
        const int pos = p0 + q;              // split: cnt_in = pos+1
        if (pos < N_ROWS - 1) {
            const float c1p = ex1 + l1[q];
            const float c2p = ex2 + l2[q];
            const float cin  = (float)(pos + 1);
            const float cout = Nf - cin;
            const float so = tot1 - c1p;
            const float qo = tot2 - c2p;
            const float within = (c2p - c1p * c1p / cin) + (qo - so * so / cout);
            const float h = within / total_scatter;
            if (h < besth) { besth = h; bestp = pos; bestc1 = c1p; }
        }
    }
    hbuf[tid] = besth; pbuf[tid] = bestp; c1buf[tid] = bestc1;
    __syncthreads();

    for (int s = 512; s > 0; s >>= 1) {
        if (tid < s) {
            const float h2 = hbuf[tid + s];
            if (h2 < hbuf[tid] || (h2 == hbuf[tid] && pbuf[tid + s] < pbuf[tid])) {
                hbuf[tid] = h2; pbuf[tid] = pbuf[tid + s]; c1buf[tid] = c1buf[tid + s];
            }
        }
        __syncthreads();
    }
    if (tid == 0)
        out[0] = c1buf[0] / (float)(pbuf[0] + 1) + LAMB * hbuf[0];
}

extern "C" void kernel_launch(void* const* d_in, const int* in_sizes, int n_in,
                              void* d_out, int out_size, void* d_ws, size_t ws_size,
                              hipStream_t stream) {
    const float* in  = (const float*)d_in[0];
    const float* tg  = (const float*)d_in[1];
    float* out = (float*)d_out;
    float* err = (float*)d_ws;   // 8192 floats of scratch

    drae_err_kernel<<<N_ROWS / 16, 256, 0, stream>>>(in, tg, err);
    drae_sort_obj_kernel<<<1, 1024, 0, stream>>>(err, out);
}